// LightSGCNNetwork_51247549776505
// MI455X (gfx1250) — compile-verified
//
#include <hip/hip_runtime.h>
#include <hip/hip_bf16.h>
#include <math.h>

typedef __attribute__((ext_vector_type(2))) float v2f;
typedef __attribute__((ext_vector_type(8))) float v8f;

#define BETA 0.3f
#define EPSN 1e-8f

// ---------------------------------------------------------------------------
// Kernel 1: z = X @ W + b via V_WMMA_F32_16X16X4_F32.
// Grid: one block (8 waves) per 16-row M tile; wave w computes N tile w (16 cols).
// ---------------------------------------------------------------------------
__global__ __launch_bounds__(256) void gemm_wmma_f32(
    const float* __restrict__ X, const float* __restrict__ W,
    const float* __restrict__ bias, float* __restrict__ Z, int K, int D) {
  const int wave = threadIdx.x >> 5;
  const int lane = threadIdx.x & 31;
  const int l15  = lane & 15;
  const int hi   = lane >> 4;
  const int m0   = blockIdx.x * 16;
  const int n0   = wave * 16;

  const float* __restrict__ xrow = X + (size_t)(m0 + l15) * K;  // A row, m = lane&15
  const float* __restrict__ wcol = W + (n0 + l15);               // B col, n = lane&15

  v8f c = {0.f, 0.f, 0.f, 0.f, 0.f, 0.f, 0.f, 0.f};
#pragma unroll 4
  for (int k0 = 0; k0 < K; k0 += 4) {
    const int kk = k0 + 2 * hi;
    v2f a;
    a.x = xrow[kk];
    a.y = xrow[kk + 1];
    v2f b;
    b.x = wcol[(size_t)kk * D];
    b.y = wcol[(size_t)(kk + 1) * D];
    c = __builtin_amdgcn_wmma_f32_16x16x4_f32(
        /*neg_a=*/false, a, /*neg_b=*/false, b,
        /*c_mod=*/(short)0, c, /*reuse_a=*/false, /*reuse_b=*/false);
  }
  const float bv = bias[n0 + l15];
#pragma unroll
  for (int r = 0; r < 8; ++r) {
    Z[(size_t)(m0 + r + 8 * hi) * D + n0 + l15] = c[r] + bv;
  }
}

// ---------------------------------------------------------------------------
// Kernel 2: per-node inverse clamped norms: ninv[n] = 1 / max(||z_n||, eps)
// One wave per node; 128 floats = one float4 per lane.
// ---------------------------------------------------------------------------
__global__ __launch_bounds__(256) void node_norms(
    const float* __restrict__ Z, float* __restrict__ ninv, int N, int D) {
  const int node = blockIdx.x * 8 + (threadIdx.x >> 5);
  const int lane = threadIdx.x & 31;
  if (node >= N) return;
  const float4* row = (const float4*)(Z + (size_t)node * D);
  float4 v = row[lane];
  float s = v.x * v.x + v.y * v.y + v.z * v.z + v.w * v.w;
#pragma unroll
  for (int off = 16; off; off >>= 1) s += __shfl_xor(s, off, 32);
  if (lane == 0) {
    float n = fmaxf(sqrtf(s), EPSN);
    ninv[node] = 1.0f / n;
  }
}

// ---------------------------------------------------------------------------
// Kernel 3: MSE edge loss (loss2). One wave per edge; t < E => positive edge.
// ---------------------------------------------------------------------------
__global__ __launch_bounds__(256) void edge_mse(
    const float* __restrict__ Z, const float* __restrict__ ninv,
    const float* __restrict__ prob, const int* __restrict__ pe,
    const int* __restrict__ ne, int E, int D, float* __restrict__ acc) {
  __shared__ float red[8];
  const int wave = threadIdx.x >> 5;
  const int lane = threadIdx.x & 31;
  const int t = blockIdx.x * 8 + wave;
  float contrib = 0.f;
  if (t < 2 * E) {
    const bool pos = t < E;
    const int e = pos ? t : t - E;
    const int* ed = pos ? pe : ne;
    const int i = ed[e];
    const int j = ed[E + e];
    const float4* zi = (const float4*)(Z + (size_t)i * D);
    const float4* zj = (const float4*)(Z + (size_t)j * D);
    float4 a = zi[lane], b = zj[lane];
    float s = a.x * b.x + a.y * b.y + a.z * b.z + a.w * b.w;
#pragma unroll
    for (int off = 16; off; off >>= 1) s += __shfl_xor(s, off, 32);
    const float cs = s * ninv[i] * ninv[j];
    const float y = BETA * cs + 0.5f * (1.0f - BETA) * (prob[i] + prob[j]);
    const float d = pos ? (1.0f - y) : y;
    contrib = 0.5f * d * d;
  }
  if (lane == 0) red[wave] = contrib;
  __syncthreads();
  if (threadIdx.x == 0) {
    float s = 0.f;
#pragma unroll
    for (int w = 0; w < 8; ++w) s += red[w];
    atomicAdd(acc, s);
  }
}

// ---------------------------------------------------------------------------
// Kernel 4: regression (cross-entropy) loss over 6E feature rows (loss1 sum).
// One wave per row, grid-stride; reg_W slice hoisted into registers per lane.
// Row r: group g = r / E selects (a, b) index pair.
// ---------------------------------------------------------------------------
__global__ __launch_bounds__(256) void edge_ce(
    const float* __restrict__ Z, const float* __restrict__ regW,
    const float* __restrict__ regb, const int* __restrict__ pe,
    const int* __restrict__ ne, const int* __restrict__ pk,
    const int* __restrict__ nk, const int* __restrict__ target,
    int E, int D, float* __restrict__ acc) {
  __shared__ float red[8];
  const int wave = threadIdx.x >> 5;
  const int lane = threadIdx.x & 31;
  // lane handles feature dims d = lane*4 .. lane*4+3 for both halves of [za;zb]
  float wa[4][3], wb[4][3];
#pragma unroll
  for (int q = 0; q < 4; ++q)
#pragma unroll
    for (int c = 0; c < 3; ++c) {
      wa[q][c] = regW[(lane * 4 + q) * 3 + c];
      wb[q][c] = regW[(128 + lane * 4 + q) * 3 + c];
    }
  const float b0 = regb[0], b1 = regb[1], b2 = regb[2];

  float lsum = 0.f;
  const long total = 6L * E;
  for (long r = (long)blockIdx.x * 8 + wave; r < total;
       r += (long)gridDim.x * 8) {
    const int g = (int)(r / E);
    const int e = (int)(r % E);
    int ia, ib;
    switch (g) {
      case 0:  ia = pe[e];     ib = pe[E + e]; break;  // pos
      case 1:  ia = ne[e];     ib = ne[E + e]; break;  // neg
      case 2:  ia = ne[e];     ib = nk[e];     break;  // surr_neg_i
      case 3:  ia = ne[E + e]; ib = nk[e];     break;  // surr_neg_j
      case 4:  ia = pe[e];     ib = pk[e];     break;  // surr_pos_i
      default: ia = pe[E + e]; ib = pk[e];     break;  // surr_pos_j
    }
    const float4* za = (const float4*)(Z + (size_t)ia * D);
    const float4* zb = (const float4*)(Z + (size_t)ib * D);
    float4 va = za[lane], vb = zb[lane];
    const float fa[4] = {va.x, va.y, va.z, va.w};
    const float fb[4] = {vb.x, vb.y, vb.z, vb.w};
    float s0 = 0.f, s1 = 0.f, s2 = 0.f;
#pragma unroll
    for (int q = 0; q < 4; ++q) {
      s0 += fa[q] * wa[q][0] + fb[q] * wb[q][0];
      s1 += fa[q] * wa[q][1] + fb[q] * wb[q][1];
      s2 += fa[q] * wa[q][2] + fb[q] * wb[q][2];
    }
#pragma unroll
    for (int off = 16; off; off >>= 1) {
      s0 += __shfl_xor(s0, off, 32);
      s1 += __shfl_xor(s1, off, 32);
      s2 += __shfl_xor(s2, off, 32);
    }
    if (lane == 0) {
      s0 += b0; s1 += b1; s2 += b2;
      const float m = fmaxf(s0, fmaxf(s1, s2));
      const float lse = m + logf(expf(s0 - m) + expf(s1 - m) + expf(s2 - m));
      const int tg = target[r];
      const float lt = (tg == 0) ? s0 : ((tg == 1) ? s1 : s2);
      lsum += lse - lt;  // = -log_softmax[target]
    }
  }
  if (lane == 0) red[wave] = lsum;
  __syncthreads();
  if (threadIdx.x == 0) {
    float s = 0.f;
#pragma unroll
    for (int w = 0; w < 8; ++w) s += red[w];
    atomicAdd(acc, s);
  }
}

// ---------------------------------------------------------------------------
// Init + finalize
// ---------------------------------------------------------------------------
__global__ void init_acc(float* acc) {
  acc[0] = 0.f;  // loss1 sum
  acc[1] = 0.f;  // loss2 sum
}

__global__ __launch_bounds__(256) void finalize(
    const float* __restrict__ Z, const float* __restrict__ acc,
    float* __restrict__ out, long nz, float inv6E) {
  const long idx = (long)blockIdx.x * blockDim.x + threadIdx.x;
  if (idx == 0) out[0] = acc[0] * inv6E + acc[1];
  for (long i = idx; i < nz; i += (long)gridDim.x * blockDim.x)
    out[1 + i] = Z[i];
}

// ---------------------------------------------------------------------------
// Launch
// ---------------------------------------------------------------------------
extern "C" void kernel_launch(void* const* d_in, const int* in_sizes, int n_in,
                              void* d_out, int out_size, void* d_ws,
                              size_t ws_size, hipStream_t stream) {
  const float* X    = (const float*)d_in[0];
  const float* Wlin = (const float*)d_in[1];
  const float* blin = (const float*)d_in[2];
  const float* regW = (const float*)d_in[3];
  const float* regb = (const float*)d_in[4];
  const float* prob = (const float*)d_in[5];
  const int*   pe   = (const int*)d_in[6];
  const int*   ne   = (const int*)d_in[7];
  const int*   pk   = (const int*)d_in[8];
  const int*   nk   = (const int*)d_in[9];
  const int*   tgt  = (const int*)d_in[10];

  const int D = in_sizes[2];              // 128
  const int K = in_sizes[1] / D;          // 512
  const int N = in_sizes[5];              // 50000
  const int E = in_sizes[8];              // 100000

  float* Z    = (float*)d_ws;                       // N*D floats (25.6 MB)
  float* ninv = Z + (size_t)N * D;                  // N floats
  float* acc  = ninv + ((N + 63) & ~63);            // 2 floats, aligned

  float* out = (float*)d_out;

  init_acc<<<1, 1, 0, stream>>>(acc);

  // GEMM: 8 waves/block, one 16-row M-tile per block; N = 8 waves * 16 cols.
  gemm_wmma_f32<<<N / 16, 256, 0, stream>>>(X, Wlin, blin, Z, K, D);

  node_norms<<<(N + 7) / 8, 256, 0, stream>>>(Z, ninv, N, D);

  edge_mse<<<(2 * E + 7) / 8, 256, 0, stream>>>(Z, ninv, prob, pe, ne, E, D,
                                                acc + 1);

  edge_ce<<<4096, 256, 0, stream>>>(Z, regW, regb, pe, ne, pk, nk, tgt, E, D,
                                    acc);

  const long nz = (long)N * D;
  finalize<<<4096, 256, 0, stream>>>(Z, acc, out, nz, 1.0f / (6.0f * E));
}